// IndividualLandmarkViT_3418793967622
// MI455X (gfx1250) — compile-verified
//
#include <hip/hip_runtime.h>
#include <hip/hip_bf16.h>

#define B_DIM   32
#define C_DIM   768
#define H_DIM   32
#define W_DIM   32
#define HW      1024
#define LNUM    9
#define NCLS    200
#define QSTR    772   // LDS row stride (772 % 64 == 4 words -> conflict-free 16-lane column reads)

typedef __attribute__((ext_vector_type(2))) float v2f;
typedef __attribute__((ext_vector_type(8))) float v8f;

// --------------------------------------------------------------------------
// Kernel 1: distance maps + softmax + eps-renorm + argmax assignment.
// One wave = one 16x16 (L x pixel) tile of ab = q * x via
// V_WMMA_F32_16X16X4_F32. q is staged in LDS padded to 16 rows (rows 9..15
// zero) so A-fragment loads are unconditional (no EXEC toggling in the hot
// loop). 2-stage software pipeline overlaps next loads with current WMMA.
// ||x||^2 per pixel comes free from the B-fragment values.
// --------------------------------------------------------------------------
__global__ __launch_bounds__(256) void k1_maps(
    const float* __restrict__ x, const float* __restrict__ q,
    const float* __restrict__ p_bias,
    float* __restrict__ out_maps, int* __restrict__ aidx, float* __restrict__ wsel)
{
    __shared__ float qlds[16 * QSTR];
    __shared__ float asq[16];

    const int tid = threadIdx.x;
    const int b   = blockIdx.y;

    // Stage q [9,768] into LDS, zero-padded to 16 rows (8 waves reuse it).
    for (int i = tid; i < 16 * C_DIM; i += 256) {
        const int l = i / C_DIM, c = i % C_DIM;
        qlds[l * QSTR + c] = (l < LNUM) ? q[l * C_DIM + c] : 0.f;
    }
    __syncthreads();
    if (tid < 16) {
        float s = 0.f;
        for (int c = 0; c < C_DIM; ++c) { const float v = qlds[tid * QSTR + c]; s += v * v; }
        asq[tid] = s;   // zero for padded rows
    }
    __syncthreads();

    const int wave = tid >> 5;
    const int lane = tid & 31;
    const int half = lane >> 4;
    const int ln   = lane & 15;
    const int tile = blockIdx.x * 8 + wave;   // 0..63 pixel tiles
    const int pix  = tile * 16 + ln;          // this lane's pixel column (N)
    const float* xb = x + (size_t)b * C_DIM * HW;

    auto load_ab = [&](int kk, v2f& af, v2f& bf) {
        af[0] = qlds[ln * QSTR + kk];
        af[1] = qlds[ln * QSTR + kk + 1];
        bf[0] = xb[(size_t)kk * HW + pix];
        bf[1] = xb[(size_t)(kk + 1) * HW + pix];
    };

    v8f acc = {};
    float sq = 0.f;
    v2f ac, bc;
    load_ab(2 * half, ac, bc);                        // prologue
    for (int k = 4; k < C_DIM; k += 4) {
        v2f an, bn;
        load_ab(k + 2 * half, an, bn);                // prefetch next K-step
        sq += bc[0] * bc[0] + bc[1] * bc[1];
        acc = __builtin_amdgcn_wmma_f32_16x16x4_f32(false, ac, false, bc,
                                                    (short)0, acc, false, false);
        ac = an; bc = bn;
    }
    sq += bc[0] * bc[0] + bc[1] * bc[1];
    acc = __builtin_amdgcn_wmma_f32_16x16x4_f32(false, ac, false, bc,
                                                (short)0, acc, false, false);
    const float bsq = sq + __shfl_xor(sq, 16);

    // Logits: acc[r] holds ab for landmark l = r (+8 on upper half), pixel pix.
    float sc[8];
    float mx = -3.4e38f;
    for (int r = 0; r < 8; ++r) {
        const int l = r + 8 * half;
        float v = -3.4e38f;
        if (l < LNUM)
            v = -(bsq - 2.f * acc[r] + asq[l]) + p_bias[l * HW + pix];
        sc[r] = v;
        mx = fmaxf(mx, v);
    }
    mx = fmaxf(mx, __shfl_xor(mx, 16));
    float es = 0.f;
    for (int r = 0; r < 8; ++r) {
        const int l = r + 8 * half;
        const float e = (l < LNUM) ? expf(sc[r] - mx) : 0.f;
        sc[r] = e;
        es += e;
    }
    es += __shfl_xor(es, 16);
    const float inv = 1.f / es;
    float s2 = 0.f;
    for (int r = 0; r < 8; ++r) {
        const int l = r + 8 * half;
        if (l < LNUM) { sc[r] = sc[r] * inv + 1e-6f; s2 += sc[r]; }
    }
    s2 += __shfl_xor(s2, 16);
    const float inv2 = 1.f / s2;

    float bestv = -1.f; int bestl = 0;
    float* om = out_maps + (size_t)b * LNUM * HW;
    for (int r = 0; r < 8; ++r) {
        const int l = r + 8 * half;
        if (l < LNUM) {
            const float mv = sc[r] * inv2;
            om[l * HW + pix] = mv;
            if (mv > bestv) { bestv = mv; bestl = l; }   // first-max tie rule
        }
    }
    const float pv = __shfl_xor(bestv, 16);
    const int   pl = __shfl_xor(bestl, 16);
    if (pv > bestv || (pv == bestv && pl < bestl)) { bestv = pv; bestl = pl; }
    if (half == 0) {
        aidx[b * HW + pix] = bestl;
        wsel[b * HW + pix] = bestv;
    }
}

// --------------------------------------------------------------------------
// Kernel 2: masked weighted pooling as WMMA GEMM:
//   pool[b][c][l] = sum_p (assign[p]==l ? w[p] : 0) * x[b][c][p]
// M = 16 channels, N = 16 landmark slots (9 valid), K = 1024 pixels.
// 2-stage pipelined; B-fragment built with branchless selects from LDS.
// --------------------------------------------------------------------------
__global__ __launch_bounds__(256) void k2_pool(
    const float* __restrict__ x, const int* __restrict__ aidx,
    const float* __restrict__ wsel, float* __restrict__ pool)
{
    __shared__ int   aL[HW];
    __shared__ float wL[HW];
    const int tid = threadIdx.x;
    const int b   = blockIdx.y;
    for (int i = tid; i < HW; i += 256) { aL[i] = aidx[b * HW + i]; wL[i] = wsel[b * HW + i]; }
    __syncthreads();

    const int wave = tid >> 5;
    const int lane = tid & 31;
    const int half = lane >> 4;
    const int ln   = lane & 15;
    const int cbase = (blockIdx.x * 8 + wave) * 16;   // 48 channel tiles
    const float* xr = x + (size_t)b * C_DIM * HW + (size_t)(cbase + ln) * HW;

    auto load_ab = [&](int kk, v2f& af, v2f& bf) {
        af[0] = xr[kk];
        af[1] = xr[kk + 1];
        const int a0 = aL[kk], a1 = aL[kk + 1];
        bf[0] = (a0 == ln) ? wL[kk]     : 0.f;
        bf[1] = (a1 == ln) ? wL[kk + 1] : 0.f;
    };

    v8f acc = {};
    v2f ac, bc;
    load_ab(2 * half, ac, bc);
    for (int k = 4; k < HW; k += 4) {
        v2f an, bn;
        load_ab(k + 2 * half, an, bn);
        acc = __builtin_amdgcn_wmma_f32_16x16x4_f32(false, ac, false, bc,
                                                    (short)0, acc, false, false);
        ac = an; bc = bn;
    }
    acc = __builtin_amdgcn_wmma_f32_16x16x4_f32(false, ac, false, bc,
                                                (short)0, acc, false, false);

    float* pb = pool + (size_t)b * C_DIM * 16;
    for (int r = 0; r < 8; ++r) {
        const int row = cbase + r + 8 * half;
        pb[row * 16 + ln] = acc[r];
    }
}

// --------------------------------------------------------------------------
// Kernel 3: soft counts (deterministic scan), mean-pool guard, LayerNorm
// over [C,L]=6912 values per batch. One block per batch image.
// --------------------------------------------------------------------------
__global__ __launch_bounds__(256) void k3_mod(
    const float* __restrict__ pool, const int* __restrict__ aidx,
    const float* __restrict__ wsel, const float* __restrict__ gamma,
    const float* __restrict__ beta, float* __restrict__ out_mod)
{
    __shared__ int   aL[HW];
    __shared__ float wL[HW];
    __shared__ float cnt[16];
    __shared__ float featL[C_DIM * LNUM];
    __shared__ float red0[8], red1[8], stat[2];

    const int tid = threadIdx.x;
    const int b   = blockIdx.x;
    for (int i = tid; i < HW; i += 256) { aL[i] = aidx[b * HW + i]; wL[i] = wsel[b * HW + i]; }
    __syncthreads();
    if (tid < 16) {                                 // deterministic (ordered) bin sums
        float s = 0.f;
        for (int p = 0; p < HW; ++p) if (aL[p] == tid) s += wL[p];
        cnt[tid] = s;
    }
    __syncthreads();

    const float* pb = pool + (size_t)b * C_DIM * 16;
    float s = 0.f, ss = 0.f;
    for (int i = tid; i < C_DIM * LNUM; i += 256) {
        const int c = i / LNUM, l = i % LNUM;
        const float cn = cnt[l];
        const float v = pb[c * 16 + l] / ((cn == 0.f) ? 1.f : cn);
        featL[i] = v;
        s += v; ss += v * v;
    }
    for (int o = 16; o > 0; o >>= 1) { s += __shfl_down(s, o); ss += __shfl_down(ss, o); }
    const int lane = tid & 31, wave = tid >> 5;
    if (lane == 0) { red0[wave] = s; red1[wave] = ss; }
    __syncthreads();
    if (tid == 0) {
        float S = 0.f, SS = 0.f;
        for (int w = 0; w < 8; ++w) { S += red0[w]; SS += red1[w]; }
        const float n = (float)(C_DIM * LNUM);
        const float mean = S / n;
        const float var  = SS / n - mean * mean;   // population variance (jnp.var)
        stat[0] = mean;
        stat[1] = rsqrtf(var + 1e-5f);
    }
    __syncthreads();
    const float mean = stat[0], rstd = stat[1];
    float* ob = out_mod + (size_t)b * C_DIM * LNUM;
    for (int i = tid; i < C_DIM * LNUM; i += 256)
        ob[i] = (featL[i] - mean) * rstd * gamma[i] + beta[i];
}

// --------------------------------------------------------------------------
// Kernel 4: scores = cls_w [200,768] x mod[b][:, :8] via WMMA.
// One wave per (class-tile, batch). Bounds handled with clamped addresses +
// branchless selects (no EXEC toggling in the K loop).
// --------------------------------------------------------------------------
__global__ __launch_bounds__(32) void k4_scores(
    const float* __restrict__ cls_w, const float* __restrict__ mod,
    float* __restrict__ scores)
{
    const int lane = threadIdx.x;
    const int half = lane >> 4;
    const int ln   = lane & 15;
    const int nbase = blockIdx.x * 16;    // 13 tiles cover 200 classes
    const int b     = blockIdx.y;
    const float* mb = mod + (size_t)b * C_DIM * LNUM;
    const int  row  = nbase + ln;
    const bool rv   = (row < NCLS);
    const bool cv   = (ln < LNUM - 1);
    const int  rowc = rv ? row : (NCLS - 1);   // clamped, always in-bounds
    const int  lnc  = cv ? ln  : (LNUM - 2);
    const float* ar = cls_w + (size_t)rowc * C_DIM;

    v8f acc = {};
    for (int k = 0; k < C_DIM; k += 4) {
        const int kk = k + 2 * half;
        v2f afrag, bfrag;
        const float a0 = ar[kk], a1 = ar[kk + 1];
        afrag[0] = rv ? a0 : 0.f;
        afrag[1] = rv ? a1 : 0.f;
        const float b0 = mb[(size_t)kk * LNUM + lnc];
        const float b1 = mb[(size_t)(kk + 1) * LNUM + lnc];
        bfrag[0] = cv ? b0 : 0.f;
        bfrag[1] = cv ? b1 : 0.f;
        acc = __builtin_amdgcn_wmma_f32_16x16x4_f32(false, afrag, false, bfrag,
                                                    (short)0, acc, false, false);
    }
    if (cv) {
        float* sb = scores + (size_t)b * NCLS * (LNUM - 1);
        for (int r = 0; r < 8; ++r) {
            const int orow = nbase + r + 8 * half;
            if (orow < NCLS) sb[orow * (LNUM - 1) + ln] = acc[r];
        }
    }
}

// --------------------------------------------------------------------------
extern "C" void kernel_launch(void* const* d_in, const int* in_sizes, int n_in,
                              void* d_out, int out_size, void* d_ws, size_t ws_size,
                              hipStream_t stream) {
    const float* x      = (const float*)d_in[0];   // [B,C,H,W]
    const float* q      = (const float*)d_in[1];   // [L,C]
    const float* p_bias = (const float*)d_in[2];   // [L,H,W]
    const float* gamma  = (const float*)d_in[3];   // [C,L]
    const float* beta   = (const float*)d_in[4];   // [C,L]
    const float* cls_w  = (const float*)d_in[5];   // [NCLS,C]

    float* out        = (float*)d_out;
    float* out_mod    = out;                                   // [B,C,L]
    float* out_maps   = out + (size_t)B_DIM * C_DIM * LNUM;    // [B,L,H,W]
    float* out_scores = out_maps + (size_t)B_DIM * LNUM * HW;  // [B,NCLS,L-1]

    int*   aidx = (int*)d_ws;                                  // [B,HW]
    float* wsel = (float*)d_ws + (size_t)B_DIM * HW;           // [B,HW]
    float* pool = (float*)d_ws + (size_t)2 * B_DIM * HW;       // [B,C,16]

    k1_maps  <<<dim3(8, B_DIM), 256, 0, stream>>>(x, q, p_bias, out_maps, aidx, wsel);
    k2_pool  <<<dim3(6, B_DIM), 256, 0, stream>>>(x, aidx, wsel, pool);
    k3_mod   <<<dim3(B_DIM),    256, 0, stream>>>(pool, aidx, wsel, gamma, beta, out_mod);
    k4_scores<<<dim3(13, B_DIM), 32, 0, stream>>>(cls_w, out_mod, out_scores);
}